// LennardJones_5695126634504
// MI455X (gfx1250) — compile-verified
//
#include <hip/hip_runtime.h>
#include <stdint.h>

#define N_PART     4096
#define NDIM       3
#define NELEM      (N_PART * NDIM)        // 12288 floats per batch
#define THREADS    256
#define JCHUNKS    4
#define CHUNK_PART (N_PART / JCHUNKS)     // 1024 j-particles per block
#define CHUNK_ELEM (CHUNK_PART * NDIM)    // 3072 floats = 12 KB per block

typedef __attribute__((ext_vector_type(4))) unsigned int v4u;
typedef __attribute__((ext_vector_type(8))) int          v8i;
typedef __attribute__((ext_vector_type(4))) int          v4i;

__global__ void lj_zero_out(float* __restrict__ out, int n) {
    int i = threadIdx.x;
    if (i < n) out[i] = 0.0f;
}

__global__ __launch_bounds__(THREADS)
void lj_energy_kernel(const float* __restrict__ x,
                      const float* __restrict__ eps_p,
                      const float* __restrict__ sig_p,
                      const float* __restrict__ box_p,
                      float* __restrict__ out)
{
    __shared__ float sx[CHUNK_ELEM];          // j-chunk, AoS [1024][3], 12 KB
    __shared__ float red[THREADS / 32];

    const int tid   = threadIdx.x;
    const int c     = blockIdx.y;             // j-chunk id
    const int b     = blockIdx.z;             // batch
    const int i     = blockIdx.x * THREADS + tid;
    const int jbase = c * CHUNK_PART;

    const float* xb = x + (size_t)b * NELEM;
    const float* xc = xb + (size_t)jbase * NDIM;

#if defined(__gfx1250__) && __has_builtin(__builtin_amdgcn_tensor_load_to_lds)
    // --- Tensor Data Mover: one 12KB DMA of this block's j-chunk into LDS ---
    if (tid == 0) {
        uint64_t ga      = (uint64_t)(uintptr_t)xc;
        uint32_t lds_off = (uint32_t)(uintptr_t)&sx[0];   // generic addr low 32 = LDS offset

        // D# group 0 (128b): count=1 | lds_addr | global_addr[56:0] | type=2
        v4u g0;
        g0.x = 1u;                                        // count=1, user mode
        g0.y = lds_off;                                   // lds_addr [63:32]
        g0.z = (uint32_t)ga;                              // global_addr [95:64]
        g0.w = (uint32_t)((ga >> 32) & 0x01FFFFFFu)       // global_addr [120:96]
             | (2u << 30);                                // type=2 ("image") [127:126]

        // D# group 1 (256b): 1-D tile of 3072 4-byte elements
        v8i g1;
        g1[0] = (int)(2u << 16);                          // data_size=2 -> 4 bytes; wg_mask=0
        g1[1] = (int)((CHUNK_ELEM & 0xFFFFu) << 16);      // tensor_dim0[15:0]  @ bits[63:48]
        g1[2] = (int)(1u << 16);                          // tensor_dim0 hi=0 | tensor_dim1=1
        g1[3] = (int)((CHUNK_ELEM & 0xFFFFu) << 16);      // tile_dim0=3072     @ bits[127:112]
        g1[4] = 1;                                        // tile_dim1=1
        g1[5] = (int)CHUNK_ELEM;                          // tensor_dim0_stride[31:0]
        g1[6] = (int)((CHUNK_ELEM & 0xFFFFu) << 16);      // dim0_stride hi=0 | dim1_stride lo
        g1[7] = 0;                                        // dim1_stride hi

        v4i g2 = {0, 0, 0, 0};                            // <=2D: groups 2/3 unused
        v4i g3 = {0, 0, 0, 0};
        v8i g4 = {0, 0, 0, 0, 0, 0, 0, 0};                // clang-23 form: extra 256b group

        __builtin_amdgcn_tensor_load_to_lds(g0, g1, g2, g3, g4, 0);
        __builtin_amdgcn_s_wait_tensorcnt(0);             // s_wait_tensorcnt 0
    }
    __syncthreads();
#else
    // Fallback: cooperative staged copy
    for (int k = tid; k < CHUNK_ELEM; k += THREADS) sx[k] = xc[k];
    __syncthreads();
#endif

    const float eps     = *eps_p;
    const float sig     = *sig_p;
    const float box     = *box_p;
    const float inv_box = 1.0f / box;     // one precise divide, hoisted
    const float sig2    = sig * sig;

    // own position straight from global (coalesced 12 B/lane; L2-resident)
    const float xi = xb[i * 3 + 0];
    const float yi = xb[i * 3 + 1];
    const float zi = xb[i * 3 + 2];

    const int iself = i - jbase;          // LDS-local index of self, or out of range

    float acc = 0.0f;
#pragma unroll 8
    for (int jj = 0; jj < CHUNK_PART; ++jj) {
        // broadcast reads: all lanes of a wave read the same LDS words
        float dx = xi - sx[jj * 3 + 0];
        float dy = yi - sx[jj * 3 + 1];
        float dz = zi - sx[jj * 3 + 2];
        // minimum image: v_rndne_f32 matches jnp.round (half-to-even)
        dx -= box * rintf(dx * inv_box);
        dy -= box * rintf(dy * inv_box);
        dz -= box * rintf(dz * inv_box);
        float r2 = dx * dx + dy * dy + dz * dz;
        r2 = (jj == iself) ? 3.0e38f : r2;        // branch-free self-exclusion
        float s2 = sig2 * __builtin_amdgcn_rcpf(r2);
        float s6 = s2 * s2 * s2;
        acc += s6 * s6 - s6;                       // fma
    }

    // wave32 shuffle reduction, then one cross-wave LDS pass
    float v = acc;
    #pragma unroll
    for (int off = 16; off > 0; off >>= 1) v += __shfl_down(v, off, 32);

    const int wave = tid >> 5, lane = tid & 31;
    if (lane == 0) red[wave] = v;
    __syncthreads();
    if (wave == 0) {
        float t = (lane < THREADS / 32) ? red[lane] : 0.0f;
        #pragma unroll
        for (int off = 4; off > 0; off >>= 1) t += __shfl_down(t, off, 32);
        // every ordered pair counted once -> i<j sum is half; 4*eps/2 = 2*eps
        if (lane == 0) atomicAdd(&out[b], 2.0f * eps * t);
    }
}

extern "C" void kernel_launch(void* const* d_in, const int* in_sizes, int n_in,
                              void* d_out, int out_size, void* d_ws, size_t ws_size,
                              hipStream_t stream) {
    const float* x   = (const float*)d_in[0];
    const float* eps = (const float*)d_in[1];
    const float* sig = (const float*)d_in[2];
    const float* box = (const float*)d_in[3];
    float*       out = (float*)d_out;

    const int B = in_sizes[0] / NELEM;   // 4

    lj_zero_out<<<1, 64, 0, stream>>>(out, out_size);

    dim3 grid(N_PART / THREADS, JCHUNKS, B);   // (16, 4, 4) = 256 blocks of 8 wave32s
    lj_energy_kernel<<<grid, THREADS, 0, stream>>>(x, eps, sig, box, out);
}